// SocialInteraction_16716012716115
// MI455X (gfx1250) — compile-verified
//
#include <hip/hip_runtime.h>
#include <math.h>

typedef __attribute__((ext_vector_type(2))) float v2f;
typedef __attribute__((ext_vector_type(8))) float v8f;

#define P_DIM 1024
#define M_DIM 64
#define R_DIM 64

// ---------------------------------------------------------------------------
// Kernel A: hi[i] = <h[i,:], Wi>,  hj[i] = <h[i,:], Wj>     (tiny, 1024 rows)
// ---------------------------------------------------------------------------
__global__ void precompute_hproj_kernel(const float* __restrict__ h,
                                        const float* __restrict__ W,
                                        float* __restrict__ hi,
                                        float* __restrict__ hj) {
    int i = blockIdx.x * blockDim.x + threadIdx.x;
    if (i >= P_DIM) return;
    const float4* hr = reinterpret_cast<const float4*>(h + (size_t)i * M_DIM);
    const float4* wi = reinterpret_cast<const float4*>(W + R_DIM);
    const float4* wj = reinterpret_cast<const float4*>(W + R_DIM + M_DIM);
    float ai = 0.f, aj = 0.f;
#pragma unroll
    for (int k = 0; k < M_DIM / 4; ++k) {
        float4 hv = hr[k];
        float4 a = wi[k];
        float4 b = wj[k];
        ai += hv.x * a.x + hv.y * a.y + hv.z * a.z + hv.w * a.w;
        aj += hv.x * b.x + hv.y * b.y + hv.z * b.z + hv.w * b.w;
    }
    hi[i] = ai;
    hj[i] = aj;
}

// ---------------------------------------------------------------------------
// Kernel B: stream rela (256 MB, the roofline term) once.
// S[i,j] = where(==0, -1e-6, where(mask, <rela[i,j,:],Wr> + hi[i]+hj[j]+b, 0))
// One thread per (i,j); 64-float dot via 16 float4 loads.
// ---------------------------------------------------------------------------
__global__ void scores_kernel(const float* __restrict__ rela,
                              const int* __restrict__ nei,
                              const float* __restrict__ W,
                              const float* __restrict__ bptr,
                              const float* __restrict__ hi,
                              const float* __restrict__ hj,
                              float* __restrict__ S) {
    __shared__ float wr[R_DIM];
    int tid = threadIdx.x;
    if (tid < R_DIM) wr[tid] = W[tid];
    __syncthreads();

    int gid = blockIdx.x * blockDim.x + tid;     // gid = i*1024 + j
    int i = gid >> 10;
    int j = gid & (P_DIM - 1);

    const float4* rp = reinterpret_cast<const float4*>(rela) + (size_t)gid * (R_DIM / 4);
    const float4* wp = reinterpret_cast<const float4*>(wr);
    float acc = 0.f;
#pragma unroll
    for (int k = 0; k < R_DIM / 4; ++k) {
        float4 rv = rp[k];
        float4 wv = wp[k];
        acc += rv.x * wv.x + rv.y * wv.y + rv.z * wv.z + rv.w * wv.w;
    }
    float s = acc + hi[i] + hj[j] + bptr[0];
    float pos = (nei[gid] > 0) ? s : 0.0f;
    if (pos == 0.0f) pos = -1e-6f;               // matches reference exactly
    S[gid] = pos;
}

// ---------------------------------------------------------------------------
// Kernel C: per block = 16 output rows.
// Phase 1 (8 waves, 2 rows each): row softmax of S, re-mask, store attn
//          transposed in LDS: attnT[j][row]  (1024*16*4 = 64 KB).
// Phase 2 (waves 0-3): out[16x64] = attn[16x1024] @ h[1024x64] with
//          V_WMMA_F32_16X16X4_F32, one 16-wide N slab per wave, K loop of 256.
// ---------------------------------------------------------------------------
__global__ void attn_wmma_kernel(const float* __restrict__ S,
                                 const int* __restrict__ nei,
                                 const float* __restrict__ h,
                                 float* __restrict__ out) {
    __shared__ float attnT[P_DIM][16];   // attnT[j][row], exactly 64 KB

    const int i0   = blockIdx.x * 16;
    const int tid  = threadIdx.x;
    const int wave = tid >> 5;
    const int lane = tid & 31;

    // ---- Phase 1: masked softmax for rows i0 .. i0+15 ----
    for (int rr = 0; rr < 2; ++rr) {
        const int row = wave * 2 + rr;
        const int i   = i0 + row;
        const float* srow = S + (size_t)i * P_DIM;

        float m = -3.402823466e+38f;
        for (int j = lane; j < P_DIM; j += 32) m = fmaxf(m, srow[j]);
        for (int off = 16; off > 0; off >>= 1) m = fmaxf(m, __shfl_xor(m, off));

        float sum = 0.f;
        for (int j = lane; j < P_DIM; j += 32) {
            float e = __expf(srow[j] - m);
            attnT[j][row] = e;
            sum += e;
        }
        for (int off = 16; off > 0; off >>= 1) sum += __shfl_xor(sum, off);
        const float inv = 1.0f / sum;

        const int* nrow = nei + (size_t)i * P_DIM;
        for (int j = lane; j < P_DIM; j += 32) {
            float a = (nrow[j] > 0) ? attnT[j][row] * inv : 0.0f;
            attnT[j][row] = a;
        }
    }
    __syncthreads();

    // ---- Phase 2: WMMA f32 16x16x4 (waves 0-3, full EXEC per wave) ----
    if (wave < 4) {
        const int n0   = wave * 16;        // N slab
        const int half = lane >> 4;        // 0: K=0,1 ; 1: K=2,3
        const int lm   = lane & 15;        // M row (A) / N col (B)
        v8f c = {};
        for (int k = 0; k < P_DIM; k += 4) {
            const int k0 = k + 2 * half;
            // A fragment (16x4): lane = M + 16*(K>=2), vgpr = K&1
            v2f a;
            a.x = attnT[k0][lm];
            a.y = attnT[k0 + 1][lm];
            // B fragment (4x16): lane = N + 16*(K>=2), vgpr = K&1
            v2f b;
            b.x = h[(size_t)k0 * M_DIM + n0 + lm];
            b.y = h[(size_t)(k0 + 1) * M_DIM + n0 + lm];
            c = __builtin_amdgcn_wmma_f32_16x16x4_f32(
                    /*neg_a=*/false, a, /*neg_b=*/false, b,
                    /*c_mod=*/(short)0, c, /*reuse_a=*/false, /*reuse_b=*/false);
        }
        // D layout: vgpr v -> M = v + 8*half, N = lm
#pragma unroll
        for (int v = 0; v < 8; ++v) {
            out[(size_t)(i0 + v + 8 * half) * M_DIM + n0 + lm] = c[v];
        }
    }
}

// ---------------------------------------------------------------------------
extern "C" void kernel_launch(void* const* d_in, const int* in_sizes, int n_in,
                              void* d_out, int out_size, void* d_ws, size_t ws_size,
                              hipStream_t stream) {
    const float* h    = (const float*)d_in[0];   // (1024, 64)
    const float* rela = (const float*)d_in[1];   // (1024, 1024, 64)
    // d_in[2] = corr_index: unused by the reference
    const int*   nei  = (const int*)d_in[3];     // (1024, 1024)
    const float* W    = (const float*)d_in[4];   // (192,)
    const float* b    = (const float*)d_in[5];   // scalar
    float* out = (float*)d_out;                  // (1024, 64)

    float* hi = (float*)d_ws;                    // 1024 floats
    float* hj = hi + P_DIM;                      // 1024 floats
    float* S  = hj + P_DIM;                      // 1024*1024 floats (4 MB)

    precompute_hproj_kernel<<<(P_DIM + 255) / 256, 256, 0, stream>>>(h, W, hi, hj);
    scores_kernel<<<(P_DIM * P_DIM) / 256, 256, 0, stream>>>(rela, nei, W, b, hi, hj, S);
    attn_wmma_kernel<<<P_DIM / 16, 256, 0, stream>>>(S, nei, h, out);
}